// PointShuffle_73478300500375
// MI455X (gfx1250) — compile-verified
//
#include <hip/hip_runtime.h>

typedef __attribute__((ext_vector_type(2))) float v2f;
typedef __attribute__((ext_vector_type(8))) float v8f;

#define B_  4
#define N_  2048
#define C_  128
#define K_  16

// Output layout (flat concat, reference return order):
//  grouped_points  : B*3*N*K   = 393216
//  grouped_features: B*256*N*K = 33554432
//  idx_full        : B*2*N*K   = 262144
#define OUT_PTS_OFF  0
#define OUT_FEAT_OFF 393216
#define OUT_IDX_OFF  (393216 + 33554432)

// ---------------------------------------------------------------------------
// Kernel 1: one wave per 16-query tile. A operand is pre-scaled by -2 and C
// is seeded with sqA[m] + sqB so V_WMMA_F32_16X16X4_F32 produces the squared
// distances directly. Selection is split across the two wave halves (half 0:
// candidates 0-7 of each tile, half 1: candidates 8-15); each half keeps a
// sorted 16-deep list via branchless shift-insert; lists merge at the end.
// ---------------------------------------------------------------------------
__global__ __launch_bounds__(32) void knn_wmma_kernel(
    const float* __restrict__ pts,
    int* __restrict__ idx_ws,
    float* __restrict__ out_idxfull)
{
    __shared__ float sqA[16];
    __shared__ float dtile[16 * 16];
    __shared__ float ml_d[16][2][16];
    __shared__ int   ml_i[16][2][16];

    const int wg   = blockIdx.x;        // 0 .. B_*(N_/16)-1  (512)
    const int b    = wg >> 7;           // / (N_/16)
    const int m0   = (wg & 127) << 4;   // query tile base
    const int lane = threadIdx.x;       // 0..31
    const int half = lane >> 4;         // 0 or 1
    const int lm   = lane & 15;

    const float* px = pts + (size_t)b * 3 * N_;
    const float* py = px + N_;
    const float* pz = py + N_;

    // squared norms of the 16 queries
    if (lane < 16) {
        float x = px[m0 + lane], y = py[m0 + lane], z = pz[m0 + lane];
        sqA[lane] = x * x + y * y + z * z;
    }
    __syncthreads();

    // per-lane copy of the 8 query norms this half's C rows need (contiguous
    // 32B run -> two ds_load_b128)
    float sqAr[8];
    {
        const float4* sa = (const float4*)&sqA[8 * half];
        float4 s0 = sa[0], s1 = sa[1];
        sqAr[0] = s0.x; sqAr[1] = s0.y; sqAr[2] = s0.z; sqAr[3] = s0.w;
        sqAr[4] = s1.x; sqAr[5] = s1.y; sqAr[6] = s1.z; sqAr[7] = s1.w;
    }

    // A operand (16x4, K=3 zero-padded), pre-scaled by -2:
    // VGPR0 = {K0 | K2}, VGPR1 = {K1 | K3}
    v2f a;
    a.x = -2.0f * (half ? pz[m0 + lm] : px[m0 + lm]);
    a.y = half ? 0.0f : (-2.0f * py[m0 + lm]);

    float best[K_]; int bid[K_];
#pragma unroll
    for (int t = 0; t < K_; ++t) { best[t] = 3.4e38f; bid[t] = 0; }

    for (int j = 0; j < N_ / 16; ++j) {
        const int n0 = j << 4;

        const float cx = px[n0 + lm];
        const float cy = py[n0 + lm];
        const float cz = pz[n0 + lm];
        const float sqB = cx * cx + cy * cy + cz * cz;

        // B operand (4x16), same striping as A (unscaled)
        v2f bv;
        bv.x = half ? cz   : cx;
        bv.y = half ? 0.0f : cy;

        // seed C with sqA[m] + sqB; wmma then yields d2 directly
        v8f c;
#pragma unroll
        for (int v = 0; v < 8; ++v) c[v] = sqAr[v] + sqB;

        // 8 args: (neg_a, A, neg_b, B, c_mod, C, reuse_a, reuse_b)
        c = __builtin_amdgcn_wmma_f32_16x16x4_f32(
                false, a, false, bv, (short)0, c, false, false);

#pragma unroll
        for (int v = 0; v < 8; ++v) {
            const int m = v + 8 * half;          // C layout: row = v (+8 on hi half)
            dtile[m * 16 + lm] = c[v];
        }
        __syncthreads();

        // lane owns query lm; this half's 8 candidates are contiguous in the
        // row -> fetch with two ds_load_b128, then fully unrolled insertion
        float cand[8];
        {
            const float4* row = (const float4*)&dtile[lm * 16 + 8 * half];
            float4 r0 = row[0], r1 = row[1];
            cand[0] = r0.x; cand[1] = r0.y; cand[2] = r0.z; cand[3] = r0.w;
            cand[4] = r1.x; cand[5] = r1.y; cand[6] = r1.z; cand[7] = r1.w;
        }
#pragma unroll
        for (int cc8 = 0; cc8 < 8; ++cc8) {
            float dd = cand[cc8];
            int   ii = n0 + 8 * half + cc8;

            // branchless sorted shift-insert (ascending; strict < keeps
            // earlier-scanned == earlier-index entries ahead on ties)
            bool lt[K_];
#pragma unroll
            for (int t = 0; t < K_; ++t) lt[t] = dd < best[t];
#pragma unroll
            for (int t = K_ - 1; t >= 1; --t) {
                best[t] = lt[t - 1] ? best[t - 1] : (lt[t] ? dd : best[t]);
                bid[t]  = lt[t - 1] ? bid[t - 1]  : (lt[t] ? ii : bid[t]);
            }
            best[0] = lt[0] ? dd : best[0];
            bid[0]  = lt[0] ? ii : bid[0];
        }
        __syncthreads();
    }

    // publish both halves' sorted lists, then merge (two-pointer, 16 steps)
#pragma unroll
    for (int t = 0; t < K_; ++t) {
        ml_d[lm][half][t] = best[t];
        ml_i[lm][half][t] = bid[t];
    }
    __syncthreads();

    if (half == 0) {
        const int q = m0 + lm;
        int*   orow = idx_ws      + ((size_t)b * N_ + q) * K_;
        float* f0   = out_idxfull + (((size_t)b * 2 + 0) * N_ + q) * K_;
        float* f1   = out_idxfull + (((size_t)b * 2 + 1) * N_ + q) * K_;

        int p0 = 0, p1 = 0;
#pragma unroll
        for (int t = 0; t < K_; ++t) {
            float d0 = ml_d[lm][0][p0], d1 = ml_d[lm][1][p1];
            int   i0 = ml_i[lm][0][p0], i1 = ml_i[lm][1][p1];
            // tie -> smaller index first (top_k stability)
            bool take0 = (d0 < d1) || (d0 == d1 && i0 < i1);
            int   oi = take0 ? i0 : i1;
            orow[t] = oi;
            f0[t]   = (float)b;
            f1[t]   = (float)oi;
            p0 += take0 ? 1 : 0;
            p1 += take0 ? 0 : 1;
        }
    }
}

// ---------------------------------------------------------------------------
// Kernel 2: grouped_points gather  (B,3,N,K) — float4 stores, int4 idx loads
// ---------------------------------------------------------------------------
__global__ void gather_points_kernel(const float* __restrict__ pts,
                                     const int*   __restrict__ idx_ws,
                                     float4*      __restrict__ out4)
{
    const int i  = blockIdx.x * blockDim.x + threadIdx.x;   // 0 .. 98303
    const int k4 = i & 3;
    const int n  = (i >> 2) & (N_ - 1);
    const int c  = (i >> 13) % 3;                            // 8192 threads per channel
    const int b  = i / 24576;                                // 3*2048*4 per batch

    const int4 id = ((const int4*)idx_ws)[((size_t)b * N_ + n) * 4 + k4];
    const float* src = pts + ((size_t)b * 3 + c) * N_;
    out4[i] = make_float4(src[id.x], src[id.y], src[id.z], src[id.w]);
}

// ---------------------------------------------------------------------------
// Kernel 3: grouped_features gather (B,256,N,K); channels 128..255 duplicate
// ---------------------------------------------------------------------------
__global__ void gather_feat_kernel(const float* __restrict__ feat,
                                   const int*   __restrict__ idx_ws,
                                   float4*      __restrict__ out4)
{
    const int i  = blockIdx.x * blockDim.x + threadIdx.x;   // 0 .. 8388607
    const int k4 = i & 3;
    const int n  = (i >> 2) & (N_ - 1);
    const int c  = (i >> 13) & 255;                          // 8192 threads per channel
    const int b  = i >> 21;                                  // 256*2048*4 per batch

    const int4 id = ((const int4*)idx_ws)[((size_t)b * N_ + n) * 4 + k4];
    const float* src = feat + ((size_t)b * C_ + (c & 127)) * N_;
    out4[i] = make_float4(src[id.x], src[id.y], src[id.z], src[id.w]);
}

// ---------------------------------------------------------------------------
extern "C" void kernel_launch(void* const* d_in, const int* in_sizes, int n_in,
                              void* d_out, int out_size, void* d_ws, size_t ws_size,
                              hipStream_t stream)
{
    const float* pts  = (const float*)d_in[0];   // (4,3,2048)
    const float* feat = (const float*)d_in[1];   // (4,128,2048)
    float* out = (float*)d_out;
    int*   idx_ws = (int*)d_ws;                  // B*N*K ints = 512 KB

    float* out_pts = out + OUT_PTS_OFF;
    float* out_ftr = out + OUT_FEAT_OFF;
    float* out_idx = out + OUT_IDX_OFF;

    // 1) KNN via WMMA distance tiles: 512 waves (one per 16-query tile)
    knn_wmma_kernel<<<B_ * (N_ / 16), 32, 0, stream>>>(pts, idx_ws, out_idx);

    // 2) grouped_points: 393216 floats / 4 per thread
    gather_points_kernel<<<(B_ * 3 * N_ * K_ / 4) / 256, 256, 0, stream>>>(
        pts, idx_ws, (float4*)out_pts);

    // 3) grouped_features (duplicated to 256 ch): 33554432 floats / 4 per thread
    gather_feat_kernel<<<(B_ * 2 * C_ * N_ * K_ / 4) / 256, 256, 0, stream>>>(
        feat, idx_ws, (float4*)out_ftr);
}